// SubstructureLevelInteractiveLigand_44779329028362
// MI455X (gfx1250) — compile-verified
//
#include <hip/hip_runtime.h>
#include <math.h>

typedef float v2f __attribute__((ext_vector_type(2)));
typedef float v8f __attribute__((ext_vector_type(8)));

#define D 128
#define NBRED 512

static __device__ __forceinline__ float sigm(float x) { return 1.0f / (1.0f + __expf(-x)); }
static __device__ __forceinline__ float leaky(float x) { return x >= 0.0f ? x : 0.2f * x; }

// D = A(16x4) x B(4x16) + C ; fp32 WMMA
static __device__ __forceinline__ v8f wmma_f32(v2f a, v2f b, v8f c) {
  return __builtin_amdgcn_wmma_f32_16x16x4_f32(false, a, false, b, (short)0, c, false, false);
}

// Load a 16x4 A-fragment from a row-major [rows x 128] matrix at (row0, k0).
// Also serves B-fragments of W^T for row-major W [cols_out x 128].
static __device__ __forceinline__ v2f frag_g(const float* __restrict__ p, int row0, int k0, int lane) {
  int mm = lane & 15, hh = lane >> 4;
  const float* q = p + ((size_t)(row0 + mm) << 7) + (k0 + (hh << 1));
  v2f r; r.x = q[0]; r.y = q[1];
  return r;
}

// ---------------- gb = column sum of Z ----------------
__global__ void __launch_bounds__(D) slil_colsum(const float* __restrict__ Z, float* __restrict__ pcol, int N) {
  int c = threadIdx.x, b = blockIdx.x;
  float s = 0.0f;
  for (int r = b; r < N; r += gridDim.x) s += Z[(size_t)r * D + c];
  pcol[b * D + c] = s;
}
__global__ void __launch_bounds__(D) slil_colsum_fin(const float* __restrict__ pcol, float* __restrict__ gb) {
  int c = threadIdx.x;
  float s = 0.0f;
  for (int b = 0; b < NBRED; ++b) s += pcol[b * D + c];
  gb[c] = s;
}

// ---------------- row norms of H ----------------
__global__ void __launch_bounds__(256) slil_rownorm(const float* __restrict__ H, float* __restrict__ Hn, int N) {
  int lane = threadIdx.x & 31;
  int gw = blockIdx.x * (blockDim.x >> 5) + (threadIdx.x >> 5);
  int nw = gridDim.x * (blockDim.x >> 5);
  for (int r = gw; r < N; r += nw) {
    const float* row = H + (size_t)r * D;
    float s = 0.0f;
#pragma unroll
    for (int j = 0; j < 4; ++j) { float x = row[lane + 32 * j]; s += x * x; }
#pragma unroll
    for (int o = 16; o > 0; o >>= 1) s += __shfl_xor(s, o, 32);
    if (lane == 0) Hn[r] = sqrtf(s);
  }
}

// ---------------- msg = H @ Wm^T + bm (WMMA, 2 row-tiles / wave, 4 waves/block) ----------------
__global__ void __launch_bounds__(128, 1) slil_msg(const float* __restrict__ H, const float* __restrict__ Wm,
                                                   const float* __restrict__ bm, float* __restrict__ msg,
                                                   int ntiles) {
  int lane = threadIdx.x & 31, wave = threadIdx.x >> 5;
  int base = (blockIdx.x * 4 + wave) * 2;
  if (base >= ntiles) return;
  bool has2 = (base + 1) < ntiles;
  int r0a = base * 16;
  int r0b = has2 ? (base + 1) * 16 : r0a;
  int m = lane & 15, hi = lane >> 4;
  v2f af0[32], af1[32];
#pragma unroll
  for (int kk = 0; kk < 32; ++kk) {
    af0[kk] = frag_g(H, r0a, kk * 4, lane);
    af1[kk] = frag_g(H, r0b, kk * 4, lane);
  }
  for (int ct = 0; ct < 8; ++ct) {
    v8f a0 = {}, a1 = {};
#pragma unroll
    for (int kk = 0; kk < 32; ++kk) {
      v2f b = frag_g(Wm, ct * 16, kk * 4, lane);
      a0 = wmma_f32(af0[kk], b, a0);
      a1 = wmma_f32(af1[kk], b, a1);
    }
    int col = ct * 16 + m;
    float bc = bm[col];
#pragma unroll
    for (int v = 0; v < 8; ++v) msg[(size_t)(r0a + v + 8 * hi) * D + col] = a0[v] + bc;
    if (has2) {
#pragma unroll
      for (int v = 0; v < 8; ++v) msg[(size_t)(r0b + v + 8 * hi) * D + col] = a1[v] + bc;
    }
  }
}

// ---------------- ||gb|| ----------------
__global__ void __launch_bounds__(32) slil_gbnorm(const float* __restrict__ gb, float* __restrict__ sc) {
  int lane = threadIdx.x;
  float s = 0.0f;
#pragma unroll
  for (int j = 0; j < 4; ++j) { float x = gb[lane + 32 * j]; s += x * x; }
#pragma unroll
  for (int o = 16; o > 0; o >>= 1) s += __shfl_xor(s, o, 32);
  if (lane == 0) sc[0] = sqrtf(s);
}

// ---------------- cos sim + block max ----------------
__global__ void __launch_bounds__(256) slil_cos(const float* __restrict__ H, const float* __restrict__ gb,
                                                const float* __restrict__ Hn, const float* __restrict__ sc,
                                                float* __restrict__ cosb, float* __restrict__ pmax, int N) {
  __shared__ float sred[256];
  int lane = threadIdx.x & 31;
  int gw = blockIdx.x * (blockDim.x >> 5) + (threadIdx.x >> 5);
  int nw = gridDim.x * (blockDim.x >> 5);
  float g0 = gb[lane], g1 = gb[lane + 32], g2 = gb[lane + 64], g3 = gb[lane + 96];
  float gbn = sc[0];
  float lmax = -3.4e38f;
  for (int r = gw; r < N; r += nw) {
    const float* row = H + (size_t)r * D;
    float s = row[lane] * g0 + row[lane + 32] * g1 + row[lane + 64] * g2 + row[lane + 96] * g3;
#pragma unroll
    for (int o = 16; o > 0; o >>= 1) s += __shfl_xor(s, o, 32);
    float cs = s / fmaxf(Hn[r] * gbn, 1e-8f);
    if (lane == 0) cosb[r] = cs;
    lmax = fmaxf(lmax, cs);
  }
  sred[threadIdx.x] = lmax;
  __syncthreads();
  for (int o = 128; o > 0; o >>= 1) {
    if ((int)threadIdx.x < o) sred[threadIdx.x] = fmaxf(sred[threadIdx.x], sred[threadIdx.x + o]);
    __syncthreads();
  }
  if (threadIdx.x == 0) pmax[blockIdx.x] = sred[0];
}

__global__ void __launch_bounds__(256) slil_maxfin(const float* __restrict__ pmax, float* __restrict__ sc) {
  __shared__ float sred[256];
  int t = threadIdx.x;
  sred[t] = fmaxf(pmax[t], pmax[t + 256]);
  __syncthreads();
  for (int o = 128; o > 0; o >>= 1) {
    if (t < o) sred[t] = fmaxf(sred[t], sred[t + o]);
    __syncthreads();
  }
  if (t == 0) sc[1] = sred[0];
}

// ---------------- exp-sum + weighted msg reduction ----------------
__global__ void __launch_bounds__(D) slil_expsum(const float* __restrict__ cosb, const float* __restrict__ msg,
                                                 const float* __restrict__ sc, float* __restrict__ ps,
                                                 float* __restrict__ pv, int N) {
  int c = threadIdx.x, b = blockIdx.x;
  float mx = sc[1];
  float vloc = 0.0f, sloc = 0.0f;
  for (int r = b; r < N; r += gridDim.x) {
    float e = expf(cosb[r] - mx);
    vloc += e * msg[(size_t)r * D + c];
    sloc += e;
  }
  pv[b * D + c] = vloc;
  if (c == 0) ps[b] = sloc;
}

// ---------------- bridge update (single block, 128 threads) ----------------
__global__ void __launch_bounds__(D) slil_bridge(const float* __restrict__ pv, const float* __restrict__ ps,
                            const float* __restrict__ Wb, const float* __restrict__ bb,
                            const float* __restrict__ Wu, const float* __restrict__ bu,
                            const float* __restrict__ Wih_b, const float* __restrict__ Whh_b,
                            const float* __restrict__ bih_b, const float* __restrict__ bhh_b,
                            const float* __restrict__ Wm, const float* __restrict__ bm,
                            float* __restrict__ gbws, float* __restrict__ ub2a_ws,
                            float* __restrict__ cu_ws) {
  __shared__ float su[D], sgbold[D], sgn[D], sgb2[D], sub[D], sred[D];
  int c = threadIdx.x;
  float vloc = 0.0f;
  for (int b = 0; b < NBRED; ++b) vloc += pv[b * D + c];
  float sl = 0.0f;
  for (int b = c; b < NBRED; b += D) sl += ps[b];
  sred[c] = sl;
  __syncthreads();
  for (int o = 64; o > 0; o >>= 1) {
    if (c < o) sred[c] += sred[c + o];
    __syncthreads();
  }
  float ua = leaky(vloc / sred[0]);                       // u_a2b
  su[c] = ua; sgbold[c] = gbws[c];
  __syncthreads();
  float d1 = 0.0f, d2 = 0.0f;
  for (int k = 0; k < D; ++k) { d1 += Wb[c * D + k] * sgbold[k]; d2 += Wu[c * D + k] * su[k]; }
  float g = sigm(d1 + bb[c] + d2 + bu[c]);
  float gn = (1.0f - g) * ua + g * sgbold[c];             // gb_new = WG(gb, u_a2b)
  sgn[c] = gn;
  __syncthreads();
  float gir = bih_b[c], giz = bih_b[c + D], gin = bih_b[c + 2 * D];
  float ghr = bhh_b[c], ghz = bhh_b[c + D], ghn = bhh_b[c + 2 * D];
  for (int k = 0; k < D; ++k) {
    float uk = su[k], gk = sgn[k];
    gir += Wih_b[c * D + k] * uk; giz += Wih_b[(c + D) * D + k] * uk; gin += Wih_b[(c + 2 * D) * D + k] * uk;
    ghr += Whh_b[c * D + k] * gk; ghz += Whh_b[(c + D) * D + k] * gk; ghn += Whh_b[(c + 2 * D) * D + k] * gk;
  }
  float r = sigm(gir + ghr), z = sigm(giz + ghz);
  float nn = tanhf(gin + r * ghn);
  float gbnew = (1.0f - z) * nn + z * gn;                 // gb = GRU_b(u_a2b, gb_new)
  gbws[c] = gbnew; sgb2[c] = gbnew;
  __syncthreads();
  float dm = bm[c];
  for (int k = 0; k < D; ++k) dm += Wm[c * D + k] * sgb2[k];
  float ub = leaky(dm);                                   // u_b2a
  ub2a_ws[c] = ub; sub[c] = ub;
  __syncthreads();
  float dc = bu[c];
  for (int k = 0; k < D; ++k) dc += Wu[c * D + k] * sub[k];
  cu_ws[c] = dc;                                          // Wu @ u_b2a + bu
}

// ---------------- big fused node update (WMMA, 2 row-tiles / wave, 4 waves/block) ----------------
__global__ void __launch_bounds__(128, 1) slil_node(float* __restrict__ Z, const float* __restrict__ Wb,
                          const float* __restrict__ bb, const float* __restrict__ Wih,
                          const float* __restrict__ Whh, const float* __restrict__ bih,
                          const float* __restrict__ bhh, const float* __restrict__ ub2a,
                          const float* __restrict__ cu, int ntiles) {
  __shared__ float smn[4 * 2 * 16 * D];                   // 64KB: per-wave two 16x128 msg_node tiles
  int lane = threadIdx.x & 31, wave = threadIdx.x >> 5;
  int base = (blockIdx.x * 4 + wave) * 2;
  if (base >= ntiles) return;
  bool has2 = (base + 1) < ntiles;
  int r0a = base * 16;
  int r0b = has2 ? (base + 1) * 16 : r0a;
  int m = lane & 15, hi = lane >> 4;
  float* buf0 = smn + wave * (2 * 16 * D);
  float* buf1 = buf0 + 16 * D;

  // keep ORIGINAL Z tiles as A-fragments in registers (in-place safety + B reuse)
  v2f az0[32], az1[32];
#pragma unroll
  for (int kk = 0; kk < 32; ++kk) {
    az0[kk] = frag_g(Z, r0a, kk * 4, lane);
    az1[kk] = frag_g(Z, r0b, kk * 4, lane);
  }

  // Phase A: g = sigmoid(Z@Wb^T + bb + (Wu@u_b2a + bu)); msg_node -> LDS
  for (int ct = 0; ct < 8; ++ct) {
    v8f a0 = {}, a1 = {};
#pragma unroll
    for (int kk = 0; kk < 32; ++kk) {
      v2f b = frag_g(Wb, ct * 16, kk * 4, lane);
      a0 = wmma_f32(az0[kk], b, a0);
      a1 = wmma_f32(az1[kk], b, a1);
    }
    int col = ct * 16 + m;
    float ubc = ub2a[col];
    float cuc = cu[col] + bb[col];
#pragma unroll
    for (int v = 0; v < 8; ++v) {
      float g0 = sigm(a0[v] + cuc);
      float g1 = sigm(a1[v] + cuc);
      float zv0 = Z[(size_t)(r0a + v + 8 * hi) * D + col];
      float zv1 = Z[(size_t)(r0b + v + 8 * hi) * D + col];
      buf0[(v + 8 * hi) * D + col] = (1.0f - g0) * ubc + g0 * zv0;
      buf1[(v + 8 * hi) * D + col] = (1.0f - g1) * ubc + g1 * zv1;
    }
  }

  // Phase B: gi = msg_node@Wih^T, gh = Z@Whh^T, GRU gates, in-place store
  for (int ct = 0; ct < 8; ++ct) {
    v8f air0 = {}, aiz0 = {}, ain0 = {}, ahr0 = {}, ahz0 = {}, ahn0 = {};
    v8f air1 = {}, aiz1 = {}, ain1 = {}, ahr1 = {}, ahz1 = {}, ahn1 = {};
#pragma unroll
    for (int kk = 0; kk < 32; ++kk) {
      int k0 = kk * 4;
      v2f bir = frag_g(Wih, ct * 16, k0, lane);
      v2f biz = frag_g(Wih, D + ct * 16, k0, lane);
      v2f bin_ = frag_g(Wih, 2 * D + ct * 16, k0, lane);
      v2f bhr = frag_g(Whh, ct * 16, k0, lane);
      v2f bhz = frag_g(Whh, D + ct * 16, k0, lane);
      v2f bhn = frag_g(Whh, 2 * D + ct * 16, k0, lane);
      v2f am0, am1;
      am0.x = buf0[m * D + k0 + 2 * hi]; am0.y = buf0[m * D + k0 + 2 * hi + 1];
      am1.x = buf1[m * D + k0 + 2 * hi]; am1.y = buf1[m * D + k0 + 2 * hi + 1];
      air0 = wmma_f32(am0, bir, air0);  air1 = wmma_f32(am1, bir, air1);
      aiz0 = wmma_f32(am0, biz, aiz0);  aiz1 = wmma_f32(am1, biz, aiz1);
      ain0 = wmma_f32(am0, bin_, ain0); ain1 = wmma_f32(am1, bin_, ain1);
      ahr0 = wmma_f32(az0[kk], bhr, ahr0); ahr1 = wmma_f32(az1[kk], bhr, ahr1);
      ahz0 = wmma_f32(az0[kk], bhz, ahz0); ahz1 = wmma_f32(az1[kk], bhz, ahz1);
      ahn0 = wmma_f32(az0[kk], bhn, ahn0); ahn1 = wmma_f32(az1[kk], bhn, ahn1);
    }
    int col = ct * 16 + m;
    float bir0 = bih[col], biz0 = bih[D + col], bin0 = bih[2 * D + col];
    float bhr0 = bhh[col], bhz0 = bhh[D + col], bhn0 = bhh[2 * D + col];
#pragma unroll
    for (int v = 0; v < 8; ++v) {
      int row = r0a + v + 8 * hi;
      float h = Z[(size_t)row * D + col];
      float r = sigm(air0[v] + bir0 + ahr0[v] + bhr0);
      float zz = sigm(aiz0[v] + biz0 + ahz0[v] + bhz0);
      float nn = tanhf(ain0[v] + bin0 + r * (ahn0[v] + bhn0));
      Z[(size_t)row * D + col] = (1.0f - zz) * nn + zz * h;
    }
    if (has2) {
#pragma unroll
      for (int v = 0; v < 8; ++v) {
        int row = r0b + v + 8 * hi;
        float h = Z[(size_t)row * D + col];
        float r = sigm(air1[v] + bir0 + ahr1[v] + bhr0);
        float zz = sigm(aiz1[v] + biz0 + ahz1[v] + bhz0);
        float nn = tanhf(ain1[v] + bin0 + r * (ahn1[v] + bhn0));
        Z[(size_t)row * D + col] = (1.0f - zz) * nn + zz * h;
      }
    }
  }
}

// ---------------- phase-2 constants: u_b2h, gi_const ----------------
__global__ void __launch_bounds__(D) slil_p2(const float* __restrict__ gbws, const float* __restrict__ Wm,
                        const float* __restrict__ bm, const float* __restrict__ Wih_n,
                        const float* __restrict__ bih_n, float* __restrict__ gic) {
  __shared__ float sgb[D], su[D];
  int c = threadIdx.x;
  sgb[c] = gbws[c];
  __syncthreads();
  float dm = bm[c];
  for (int k = 0; k < D; ++k) dm += Wm[c * D + k] * sgb[k];
  su[c] = leaky(dm);
  __syncthreads();
  float a = bih_n[c], b2 = bih_n[c + D], d3 = bih_n[c + 2 * D];
  for (int k = 0; k < D; ++k) {
    float u = su[k];
    a += Wih_n[c * D + k] * u; b2 += Wih_n[(c + D) * D + k] * u; d3 += Wih_n[(c + 2 * D) * D + k] * u;
  }
  gic[c] = a; gic[c + D] = b2; gic[c + 2 * D] = d3;
}

// ---------------- H update (WMMA, constant gi, 2 row-tiles / wave, 4 waves/block) ----------------
__global__ void __launch_bounds__(128, 1) slil_hstep(float* __restrict__ H, const float* __restrict__ Whh,
                           const float* __restrict__ bhh, const float* __restrict__ gic,
                           int ntiles) {
  int lane = threadIdx.x & 31, wave = threadIdx.x >> 5;
  int base = (blockIdx.x * 4 + wave) * 2;
  if (base >= ntiles) return;
  bool has2 = (base + 1) < ntiles;
  int r0a = base * 16;
  int r0b = has2 ? (base + 1) * 16 : r0a;
  int m = lane & 15, hi = lane >> 4;
  v2f ah0[32], ah1[32];
#pragma unroll
  for (int kk = 0; kk < 32; ++kk) {
    ah0[kk] = frag_g(H, r0a, kk * 4, lane);
    ah1[kk] = frag_g(H, r0b, kk * 4, lane);
  }
  for (int ct = 0; ct < 8; ++ct) {
    v8f ar0 = {}, az0 = {}, an0 = {}, ar1 = {}, az1 = {}, an1 = {};
#pragma unroll
    for (int kk = 0; kk < 32; ++kk) {
      int k0 = kk * 4;
      v2f br = frag_g(Whh, ct * 16, k0, lane);
      v2f bz = frag_g(Whh, D + ct * 16, k0, lane);
      v2f bn = frag_g(Whh, 2 * D + ct * 16, k0, lane);
      ar0 = wmma_f32(ah0[kk], br, ar0); ar1 = wmma_f32(ah1[kk], br, ar1);
      az0 = wmma_f32(ah0[kk], bz, az0); az1 = wmma_f32(ah1[kk], bz, az1);
      an0 = wmma_f32(ah0[kk], bn, an0); an1 = wmma_f32(ah1[kk], bn, an1);
    }
    int col = ct * 16 + m;
    float gr = gic[col], gz = gic[D + col], gn = gic[2 * D + col];
    float bhr0 = bhh[col], bhz0 = bhh[D + col], bhn0 = bhh[2 * D + col];
#pragma unroll
    for (int v = 0; v < 8; ++v) {
      int row = r0a + v + 8 * hi;
      float h = H[(size_t)row * D + col];
      float r = sigm(gr + ar0[v] + bhr0);
      float zz = sigm(gz + az0[v] + bhz0);
      float nn = tanhf(gn + r * (an0[v] + bhn0));
      H[(size_t)row * D + col] = (1.0f - zz) * nn + zz * h;
    }
    if (has2) {
#pragma unroll
      for (int v = 0; v < 8; ++v) {
        int row = r0b + v + 8 * hi;
        float h = H[(size_t)row * D + col];
        float r = sigm(gr + ar1[v] + bhr0);
        float zz = sigm(gz + az1[v] + bhz0);
        float nn = tanhf(gn + r * (an1[v] + bhn0));
        H[(size_t)row * D + col] = (1.0f - zz) * nn + zz * h;
      }
    }
  }
}

extern "C" void kernel_launch(void* const* d_in, const int* in_sizes, int n_in,
                              void* d_out, int out_size, void* d_ws, size_t ws_size,
                              hipStream_t stream) {
  (void)n_in; (void)out_size; (void)ws_size;
  const float* Hin   = (const float*)d_in[0];
  const float* Zin   = (const float*)d_in[1];
  const float* Wm    = (const float*)d_in[2];
  const float* bm    = (const float*)d_in[3];
  const float* Wb    = (const float*)d_in[4];
  const float* bb    = (const float*)d_in[5];
  const float* Wu    = (const float*)d_in[6];
  const float* bu    = (const float*)d_in[7];
  const float* Wih_b = (const float*)d_in[8];
  const float* Whh_b = (const float*)d_in[9];
  const float* bih_b = (const float*)d_in[10];
  const float* bhh_b = (const float*)d_in[11];
  const float* Wih_n = (const float*)d_in[12];
  const float* Whh_n = (const float*)d_in[13];
  const float* bih_n = (const float*)d_in[14];
  const float* bhh_n = (const float*)d_in[15];

  const int N = in_sizes[0] / D;
  const int ntiles = (N + 15) / 16;
  const int nblk8 = (ntiles + 7) / 8;        // 4 waves * 2 tiles per 128-thread block

  float* Zw = (float*)d_out;                 // output Z doubles as working buffer
  float* Hw = Zw + (size_t)N * D;            // output H doubles as working buffer

  float* w     = (float*)d_ws;
  float* msg   = w;                          // N*128
  float* cosb  = msg + (size_t)N * D;        // N
  float* Hn    = cosb + N;                   // N
  float* pmax  = Hn + N;                     // 512
  float* ps    = pmax + NBRED;               // 512
  float* pv    = ps + NBRED;                 // 512*128
  float* pcol  = pv + (size_t)NBRED * D;     // 512*128
  float* gb    = pcol + (size_t)NBRED * D;   // 128
  float* sc    = gb + D;                     // scalars: [0]=||gb||, [1]=max
  float* ub2a  = sc + 8;                     // 128
  float* cu    = ub2a + D;                   // 128
  float* gic   = cu + D;                     // 384

  hipMemcpyAsync(Zw, Zin, (size_t)N * D * sizeof(float), hipMemcpyDeviceToDevice, stream);
  hipMemcpyAsync(Hw, Hin, (size_t)N * D * sizeof(float), hipMemcpyDeviceToDevice, stream);

  slil_colsum<<<NBRED, D, 0, stream>>>(Zin, pcol, N);
  slil_colsum_fin<<<1, D, 0, stream>>>(pcol, gb);
  slil_rownorm<<<NBRED, 256, 0, stream>>>(Hin, Hn, N);
  slil_msg<<<nblk8, 128, 0, stream>>>(Hin, Wm, bm, msg, ntiles);

  for (int it = 0; it < 3; ++it) {
    slil_gbnorm<<<1, 32, 0, stream>>>(gb, sc);
    slil_cos<<<NBRED, 256, 0, stream>>>(Hin, gb, Hn, sc, cosb, pmax, N);
    slil_maxfin<<<1, 256, 0, stream>>>(pmax, sc);
    slil_expsum<<<NBRED, D, 0, stream>>>(cosb, msg, sc, ps, pv, N);
    slil_bridge<<<1, D, 0, stream>>>(pv, ps, Wb, bb, Wu, bu, Wih_b, Whh_b, bih_b, bhh_b,
                                     Wm, bm, gb, ub2a, cu);
    slil_node<<<nblk8, 128, 0, stream>>>(Zw, Wb, bb, Wih_n, Whh_n, bih_n, bhh_n, ub2a, cu, ntiles);
  }

  slil_p2<<<1, D, 0, stream>>>(gb, Wm, bm, Wih_n, bih_n, gic);
  for (int it = 0; it < 3; ++it)
    slil_hstep<<<nblk8, 128, 0, stream>>>(Hw, Whh_n, bhh_n, gic, ntiles);
}